// LowRankGlobalAttention_7069516169377
// MI455X (gfx1250) — compile-verified
//
#include <hip/hip_runtime.h>

// MI455X / gfx1250: wave32, WMMA. Problem is HBM-bandwidth bound
// (512MB of x @ 23.3TB/s ~= 22us vs 32.8 GFLOP), so stay in fp32 and use
// V_WMMA_F32_16X16X4_F32 (no precision-conversion traffic).

typedef float v2f __attribute__((ext_vector_type(2)));
typedef float v8f __attribute__((ext_vector_type(8)));

#define KDIM 256   // d
#define CDIM 128   // 4*K
#define EPSF 1e-6f

// ws float layout:
//   [0..31]    u_sum  (colsum of U)
//   [32..63]   s      (colsum of V)
//   [64..1087] VtZ    (32x32 row-major, VtZ[kv*32+kz])
//   [1088]     D
#define WS_VTZ 64
#define WS_D   1088
#define WS_TOT 1089

__global__ void zero_ws(float* __restrict__ w) {
    int i = blockIdx.x * blockDim.x + threadIdx.x;
    if (i < WS_TOT) w[i] = 0.0f;
}

// ---------------------------------------------------------------------------
// Phase 1: X = relu(x@W + b).  Per block: 4 waves, 4 iterations of 64 rows
// (256 rows/block).  W staged in LDS with k-pair interleave so each B
// fragment is one 8-byte ds load (compiler fuses pairs into ds_load_2addr).
// U -> out[:,0:32] (scratch for phase 3), T -> out[:,32:64], U/V/Z staged in
// LDS for colsums + VtZ outer-product partials (global fire-and-forget
// f32 atomics: ~2M total spread over 64 cachelines -> well under L2 rate).
// ---------------------------------------------------------------------------
__global__ __launch_bounds__(128) void gemm_phase(
    const float* __restrict__ x, const float* __restrict__ Wm,
    const float* __restrict__ bv, float* __restrict__ out,
    float* __restrict__ ws, int n)
{
    extern __shared__ float smem[];
    float* Wl    = smem;          // 32768 floats (128KB), interleaved pairs
    float* stage = smem + 32768;  // 64 rows x 96 cols (U,V,Z) = 24KB

    const int tid    = threadIdx.x;
    const int wave   = tid >> 5;
    const int lane   = tid & 31;
    const int lane15 = lane & 15;
    const int koff   = (lane < 16) ? 0 : 2;   // K sub-offset (A/B layout)
    const int moff   = (lane < 16) ? 0 : 8;   // M sub-offset (C/D layout)

    // Stage W: Wl[(k>>1)*256 + c*2 + (k&1)] = W[k*128+c]
    for (int i = tid; i < KDIM * CDIM; i += 128) {
        int k = i >> 7, c = i & 127;
        Wl[(k >> 1) * 256 + (c << 1) + (k & 1)] = Wm[i];
    }
    __syncthreads();

    // Loop-invariant bias fragment (col = t*16 + lane15)
    float bias[8];
#pragma unroll
    for (int t = 0; t < 8; ++t) bias[t] = bv[t * 16 + lane15];

    // Per-thread reduction state (accumulated across the 4 row-iterations)
    float vtz[8];
#pragma unroll
    for (int e = 0; e < 8; ++e) vtz[e] = 0.0f;
    float colsum = 0.0f;                 // threads 0..63 own one of 64 cols
    const int kv  = tid >> 2;            // 0..31
    const int kzb = (tid & 3) << 3;      // 0,8,16,24

    const long long blk_row0 = (long long)blockIdx.x * 256;
    // Uniform (scalar) predicate: whole block in range -> branch-free stores.
    const bool full_block = (blk_row0 + 256 <= (long long)n);

    for (int it = 0; it < 4; ++it) {
        const int r0 = (int)blk_row0 + it * 64 + wave * 16;

        int arow = r0 + lane15;
        if (arow >= n) arow = n - 1;                 // clamp; masked later
        const float* xrow = x + (long long)arow * KDIM;

        v8f acc[8];
#pragma unroll
        for (int t = 0; t < 8; ++t)
#pragma unroll
            for (int r = 0; r < 8; ++r) acc[t][r] = 0.0f;

#pragma unroll 2
        for (int k = 0; k < KDIM; k += 4) {
            v2f a = *(const v2f*)(xrow + k + koff);          // global_load_b64
            const float* wk = Wl + ((k + koff) >> 1) * 256;
            // Issue all 8 B-fragment loads first (one dscnt wait), then WMMAs.
            v2f bb[8];
#pragma unroll
            for (int t = 0; t < 8; ++t)
                bb[t] = *(const v2f*)(wk + ((t * 16 + lane15) << 1));
#pragma unroll
            for (int t = 0; t < 8; ++t)
                acc[t] = __builtin_amdgcn_wmma_f32_16x16x4_f32(
                    false, a, false, bb[t], (short)0, acc[t], false, false);
        }

        __syncthreads();   // previous iteration's stage fully consumed

        // bias + relu + scatter: U/T to out, U/V/Z to LDS stage
        if (full_block) {
            // One base address per tile; rows become immediate offsets (r*256B).
            float* const obase = out + (long long)(r0 + moff) * 64 + lane15;
            float* const sbase = stage + (wave * 16 + moff) * 96 + lane15;
#pragma unroll
            for (int t = 0; t < 8; ++t) {
#pragma unroll
                for (int r = 0; r < 8; ++r) {
                    float v = fmaxf(acc[t][r] + bias[t], 0.0f);
                    if (t < 6)
                        sbase[r * 96 + t * 16] = v;
                    if (t < 2)
                        obase[r * 64 + t * 16] = v;           // U -> cols 0..31
                    else if (t >= 6)
                        obase[r * 64 + (t - 6) * 16 + 32] = v; // T -> cols 32..63
                }
            }
        } else {
#pragma unroll
            for (int t = 0; t < 8; ++t) {
                const int col = t * 16 + lane15;
#pragma unroll
                for (int r = 0; r < 8; ++r) {
                    const int erow  = r0 + r + moff;
                    const bool valid = (erow < n);
                    float v = fmaxf(acc[t][r] + bias[t], 0.0f);
                    if (t < 6)
                        stage[(wave * 16 + r + moff) * 96 + col] = valid ? v : 0.0f;
                    if (valid) {
                        if (t < 2)        out[(long long)erow * 64 + col]        = v;
                        else if (t >= 6)  out[(long long)erow * 64 + (col - 64)] = v;
                    }
                }
            }
        }
        __syncthreads();   // stage complete

        // colsums of U (cols 0..31 -> u_sum) and V (cols 32..63 -> s)
        if (tid < 64) {
#pragma unroll 4
            for (int lr = 0; lr < 64; ++lr) colsum += stage[lr * 96 + tid];
        }
        // VtZ partial: thread owns (kv, kzb..kzb+7)
        for (int lr = 0; lr < 64; ++lr) {
            const float v = stage[lr * 96 + 32 + kv];
            const float4 z0 = *(const float4*)(stage + lr * 96 + 64 + kzb);
            const float4 z1 = *(const float4*)(stage + lr * 96 + 64 + kzb + 4);
            vtz[0] += v * z0.x; vtz[1] += v * z0.y;
            vtz[2] += v * z0.z; vtz[3] += v * z0.w;
            vtz[4] += v * z1.x; vtz[5] += v * z1.y;
            vtz[6] += v * z1.z; vtz[7] += v * z1.w;
        }
    }

    // Fire-and-forget global float atomics (no return -> STOREcnt only)
    if (tid < 64) atomicAdd(&ws[tid], colsum);
#pragma unroll
    for (int e = 0; e < 8; ++e)
        atomicAdd(&ws[WS_VTZ + kv * 32 + kzb + e], vtz[e]);
}

// ---------------------------------------------------------------------------
// Phase 2: D = 1 / (dot(u_sum, s)/N + eps)   (trivial)
// ---------------------------------------------------------------------------
__global__ void compute_D(float* __restrict__ ws, int n) {
    if (threadIdx.x == 0 && blockIdx.x == 0) {
        float dot = 0.0f;
        for (int k = 0; k < 32; ++k) dot += ws[k] * ws[32 + k];
        ws[WS_D] = 1.0f / (dot / (float)n + EPSF);
    }
}

// ---------------------------------------------------------------------------
// Phase 3: res = (U @ VtZ) * D, in place over out[:,0:32].
// 4 waves/block, 16 rows/wave; VtZ (4KB) interleaved in LDS.
// ---------------------------------------------------------------------------
__global__ __launch_bounds__(128) void apply_phase(
    float* __restrict__ out, const float* __restrict__ ws, int n)
{
    __shared__ float Vl[1024];
    const int tid    = threadIdx.x;
    const int wave   = tid >> 5;
    const int lane   = tid & 31;
    const int lane15 = lane & 15;
    const int koff   = (lane < 16) ? 0 : 2;
    const int moff   = (lane < 16) ? 0 : 8;

    for (int i = tid; i < 1024; i += 128) {
        int k = i >> 5, j = i & 31;
        Vl[(k >> 1) * 64 + (j << 1) + (k & 1)] = ws[WS_VTZ + i];
    }
    __syncthreads();
    const float D = ws[WS_D];

    const int r0 = blockIdx.x * 64 + wave * 16;
    int arow = r0 + lane15;
    if (arow >= n) arow = n - 1;
    const float* urow = out + (long long)arow * 64;   // U lives in cols 0..31

    v8f acc[2];
#pragma unroll
    for (int t = 0; t < 2; ++t)
#pragma unroll
        for (int r = 0; r < 8; ++r) acc[t][r] = 0.0f;

#pragma unroll
    for (int k = 0; k < 32; k += 4) {
        v2f a = *(const v2f*)(urow + k + koff);
        const float* wk = Vl + ((k + koff) >> 1) * 64;
        v2f b0 = *(const v2f*)(wk + (lane15 << 1));
        v2f b1 = *(const v2f*)(wk + ((16 + lane15) << 1));
        acc[0] = __builtin_amdgcn_wmma_f32_16x16x4_f32(
            false, a, false, b0, (short)0, acc[0], false, false);
        acc[1] = __builtin_amdgcn_wmma_f32_16x16x4_f32(
            false, a, false, b1, (short)0, acc[1], false, false);
    }

    const bool full_block = ((blockIdx.x + 1) * 64 <= n);   // uniform branch
    if (full_block) {
        float* const obase = out + (long long)(r0 + moff) * 64 + lane15;
#pragma unroll
        for (int t = 0; t < 2; ++t)
#pragma unroll
            for (int r = 0; r < 8; ++r)
                obase[r * 64 + t * 16] = acc[t][r] * D;
    } else {
#pragma unroll
        for (int t = 0; t < 2; ++t)
#pragma unroll
            for (int r = 0; r < 8; ++r) {
                const int erow = r0 + r + moff;
                if (erow < n)
                    out[(long long)erow * 64 + t * 16 + lane15] = acc[t][r] * D;
            }
    }
}

extern "C" void kernel_launch(void* const* d_in, const int* in_sizes, int n_in,
                              void* d_out, int out_size, void* d_ws, size_t ws_size,
                              hipStream_t stream) {
    const float* x  = (const float*)d_in[0];
    const float* Wm = (const float*)d_in[1];
    const float* bv = (const float*)d_in[2];
    float* out = (float*)d_out;
    float* ws  = (float*)d_ws;
    const int n = in_sizes[0] / KDIM;    // N = 500000

    zero_ws<<<(WS_TOT + 255) / 256, 256, 0, stream>>>(ws);

    const int gblocks = (n + 255) / 256;
    const size_t smem = (size_t)(32768 + 64 * 96) * sizeof(float); // 152KB
    gemm_phase<<<gblocks, 128, smem, stream>>>(x, Wm, bv, out, ws, n);

    compute_D<<<1, 32, 0, stream>>>(ws, n);

    apply_phase<<<(n + 63) / 64, 128, 0, stream>>>(out, ws, n);
}